// VoLayer_38740605010354
// MI455X (gfx1250) — compile-verified
//
#include <hip/hip_runtime.h>

typedef __attribute__((ext_vector_type(2))) float v2f;
typedef __attribute__((ext_vector_type(8))) float v8f;

#define KDIM 4096
#define TPB  256
#define ROWSTRIDE 260   // words; row base 8B-aligned, b64 loads hit all 64 banks once

// 2x2 complex matrix as separate re/im, row-major index i*2+j
struct C22 { float r[4]; float i[4]; };

// C = A * B
__device__ __forceinline__ C22 cmul(const C22& A, const C22& B) {
  C22 C;
#pragma unroll
  for (int i = 0; i < 2; ++i)
#pragma unroll
    for (int j = 0; j < 2; ++j) {
      float re = 0.f, im = 0.f;
#pragma unroll
      for (int t = 0; t < 2; ++t) {
        re += A.r[i*2+t]*B.r[t*2+j] - A.i[i*2+t]*B.i[t*2+j];
        im += A.r[i*2+t]*B.i[t*2+j] + A.i[i*2+t]*B.r[t*2+j];
      }
      C.r[i*2+j] = re; C.i[i*2+j] = im;
    }
  return C;
}

// C = A * B^H   (B^H[t][j] = conj(B[j][t]))
__device__ __forceinline__ C22 cmulBh(const C22& A, const C22& B) {
  C22 C;
#pragma unroll
  for (int i = 0; i < 2; ++i)
#pragma unroll
    for (int j = 0; j < 2; ++j) {
      float re = 0.f, im = 0.f;
#pragma unroll
      for (int t = 0; t < 2; ++t) {
        re += A.r[i*2+t]*B.r[j*2+t] + A.i[i*2+t]*B.i[j*2+t];
        im += A.i[i*2+t]*B.r[j*2+t] - A.r[i*2+t]*B.i[j*2+t];
      }
      C.r[i*2+j] = re; C.i[i*2+j] = im;
    }
  return C;
}

// C = A^H * B   (A^H[i][t] = conj(A[t][i]))
__device__ __forceinline__ C22 cmulAh(const C22& A, const C22& B) {
  C22 C;
#pragma unroll
  for (int i = 0; i < 2; ++i)
#pragma unroll
    for (int j = 0; j < 2; ++j) {
      float re = 0.f, im = 0.f;
#pragma unroll
      for (int t = 0; t < 2; ++t) {
        re += A.r[t*2+i]*B.r[t*2+j] + A.i[t*2+i]*B.i[t*2+j];
        im += A.r[t*2+i]*B.i[t*2+j] - A.i[t*2+i]*B.r[t*2+j];
      }
      C.r[i*2+j] = re; C.i[i*2+j] = im;
    }
  return C;
}

__global__ __launch_bounds__(TPB)
void vo_layer_kernel(const float* __restrict__ uir, const float* __restrict__ uii,
                     const float* __restrict__ ucr, const float* __restrict__ uci,
                     const float* __restrict__ w,
                     const float* __restrict__ o_re, const float* __restrict__ o_im,
                     float* __restrict__ out)
{
  // 16 rows x 256 cols (stride 260 words). Rows 0-7 hold the 8 partial-sum
  // components (re/im interleaved); rows 8-15 are explicit zero padding so the
  // WMMA A-tile fetches need no predication (EXEC stays clean, loads are b64).
  __shared__ float lds[16 * ROWSTRIDE];

  const int b = blockIdx.x;
  const int t = threadIdx.x;

  // Zero the padding rows (8..15) once.
#pragma unroll
  for (int z = t; z < 8 * ROWSTRIDE; z += TPB) lds[8 * ROWSTRIDE + z] = 0.f;

  C22 O;
#pragma unroll
  for (int e = 0; e < 4; ++e) { O.r[e] = o_re[e]; O.i[e] = o_im[e]; }

  const size_t base = (size_t)b * KDIM;
  const float4* __restrict__ pUIr = (const float4*)uir + base;
  const float4* __restrict__ pUIi = (const float4*)uii + base;
  const float4* __restrict__ pUCr = (const float4*)ucr + base;
  const float4* __restrict__ pUCi = (const float4*)uci + base;
  const float*  __restrict__ pW   = w + base;

  // acc[2e] = Re(VO[e]), acc[2e+1] = Im(VO[e]) -> matches complex64 memory order
  float acc[8];
#pragma unroll
  for (int m = 0; m < 8; ++m) acc[m] = 0.f;

  // Streaming phase: fully coalesced b128 loads, 16 iterations per thread.
  for (int k = t; k < KDIM; k += TPB) {
    const float4 ar = pUIr[k];
    const float4 ai = pUIi[k];
    const float4 cr = pUCr[k];
    const float4 ci = pUCi[k];
    const float  ww = pW[k];

    C22 UI; UI.r[0]=ar.x; UI.r[1]=ar.y; UI.r[2]=ar.z; UI.r[3]=ar.w;
            UI.i[0]=ai.x; UI.i[1]=ai.y; UI.i[2]=ai.z; UI.i[3]=ai.w;
    C22 Uc; Uc.r[0]=cr.x; Uc.r[1]=cr.y; Uc.r[2]=cr.z; Uc.r[3]=cr.w;
            Uc.i[0]=ci.x; Uc.i[1]=ci.y; Uc.i[2]=ci.z; Uc.i[3]=ci.w;

    C22 M1 = cmulBh(UI, Uc);   // UI @ Uc^H
    C22 U  = cmul (Uc, M1);    // UI_tilde = Uc @ UI @ Uc^H
    C22 M2 = cmul (O,  U);     // O @ UI_tilde
    C22 T  = cmulAh(U, M2);    // UI_tilde^H @ O @ UI_tilde
    C22 V  = cmul (O,  T);     // VO

#pragma unroll
    for (int e = 0; e < 4; ++e) {
      acc[2*e]   = fmaf(ww, V.r[e], acc[2*e]);
      acc[2*e+1] = fmaf(ww, V.i[e], acc[2*e+1]);
    }
  }

#pragma unroll
  for (int m = 0; m < 8; ++m) lds[m * ROWSTRIDE + t] = acc[m];
  __syncthreads();

  // WMMA reduction (wave 0 only -> EXEC all ones inside this branch).
  // D = A(16x4) x ones(4x16) + C row-sums 4 columns per wmma. 64 wmmas reduce
  // all 256 thread-partials; split over 4 independent C chains for XDL ILP.
  if (t < 32) {
    const int m   = t & 15;
    const int kof = (t >> 4) << 1;          // lanes 0-15: K=0,1 ; lanes 16-31: K=2,3
    const float* __restrict__ rowp = &lds[m * ROWSTRIDE + kof];

    v2f bones; bones.x = 1.f; bones.y = 1.f;
    v8f c0 = {}, c1 = {}, c2 = {}, c3 = {};

#pragma unroll 4
    for (int q = 0; q < 16; ++q) {
      const v2f a0 = *(const v2f*)(rowp + (q * 16 + 0));
      const v2f a1 = *(const v2f*)(rowp + (q * 16 + 4));
      const v2f a2 = *(const v2f*)(rowp + (q * 16 + 8));
      const v2f a3 = *(const v2f*)(rowp + (q * 16 + 12));
      c0 = __builtin_amdgcn_wmma_f32_16x16x4_f32(false, a0, false, bones, (short)0, c0, false, false);
      c1 = __builtin_amdgcn_wmma_f32_16x16x4_f32(false, a1, false, bones, (short)0, c1, false, false);
      c2 = __builtin_amdgcn_wmma_f32_16x16x4_f32(false, a2, false, bones, (short)0, c2, false, false);
      c3 = __builtin_amdgcn_wmma_f32_16x16x4_f32(false, a3, false, bones, (short)0, c3, false, false);
    }

    const v8f csum = (c0 + c1) + (c2 + c3);

    // C/D layout: lane 0 holds D[m][0] in VGPR m for m=0..7 -> the 8 output floats.
    if (t == 0) {
      float* o = out + (size_t)b * 8;
#pragma unroll
      for (int mm = 0; mm < 8; ++mm) o[mm] = csum[mm];
    }
  }
}

extern "C" void kernel_launch(void* const* d_in, const int* in_sizes, int n_in,
                              void* d_out, int out_size, void* d_ws, size_t ws_size,
                              hipStream_t stream) {
  const float* uir = (const float*)d_in[0];
  const float* uii = (const float*)d_in[1];
  const float* ucr = (const float*)d_in[2];
  const float* uci = (const float*)d_in[3];
  const float* w   = (const float*)d_in[4];
  const float* ore = (const float*)d_in[5];
  const float* oim = (const float*)d_in[6];
  float* out = (float*)d_out;

  const int Bdim = in_sizes[4] / KDIM;   // w has B*K elements
  vo_layer_kernel<<<Bdim, TPB, 0, stream>>>(uir, uii, ucr, uci, w, ore, oim, out);
}